// PKM_7060926235264
// MI455X (gfx1250) — compile-verified
//
#include <hip/hip_runtime.h>
#include <math.h>

typedef __attribute__((ext_vector_type(2))) float v2f;
typedef __attribute__((ext_vector_type(8))) float v8f;

#define DIMX     512
#define HEADS    4
#define NUM_KEYS 256
#define TOPK     32
#define DH       128      // DIM_HEAD / 2
#define BT       4096     // B * T
#define DIMQ     1024     // DIM_QUERY

static __device__ __forceinline__ v8f wmma_f32_16x16x4(v2f a, v2f b, v8f c) {
  // D = A(16x4,f32) * B(4x16,f32) + C(16x16,f32)
  return __builtin_amdgcn_wmma_f32_16x16x4_f32(false, a, false, b, (short)0, c,
                                               false, false);
}

static __device__ __forceinline__ void wave_argmax(float& v, int& k) {
#pragma unroll
  for (int off = 16; off > 0; off >>= 1) {
    float ov = __shfl_xor(v, off, 32);
    int   ok = __shfl_xor(k, off, 32);
    if (ov > v || (ov == v && ok < k)) { v = ov; k = ok; }
  }
}

// ---------------------------------------------------------------- q = x * wq^T
__global__ void k_qgemm(const float* __restrict__ x, const float* __restrict__ wq,
                        float* __restrict__ q) {
  const int lane = threadIdx.x & 31;
  const int wid  = threadIdx.x >> 5;
  const int m0 = blockIdx.y * 16;                 // token tile
  const int n0 = (blockIdx.x * 8 + wid) * 16;     // query-dim tile
  const int r  = lane & 15;
  const int kh = (lane >> 4) * 2;                 // k-pair select (0 or 2)
  const float* ap = x  + (size_t)(m0 + r) * DIMX + kh;
  const float* bp = wq + (size_t)(n0 + r) * DIMX + kh;
  v8f c = {};
#pragma unroll 4
  for (int k = 0; k < DIMX; k += 4) {
    v2f a = *(const v2f*)ap;
    v2f b = *(const v2f*)bp;
    c = wmma_f32_16x16x4(a, b, c);
    ap += 4; bp += 4;
  }
  const int crow = m0 + ((lane >> 4) << 3);
  const int col  = n0 + (lane & 15);
#pragma unroll
  for (int i = 0; i < 8; ++i) q[(size_t)(crow + i) * DIMQ + col] = c[i];
}

// ------------------------------------------------------- batchnorm statistics
__global__ void k_bnstat(const float* __restrict__ q, float* __restrict__ psum,
                         float* __restrict__ psq) {
  const int c  = blockIdx.x * 256 + threadIdx.x;  // gridDim.x = 4
  const int r0 = blockIdx.y * 256;                // gridDim.y = 16
  float s = 0.f, ss = 0.f;
  for (int r = r0; r < r0 + 256; ++r) {
    float v = q[(size_t)r * DIMQ + c];
    s += v; ss += v * v;
  }
  psum[blockIdx.y * DIMQ + c] = s;
  psq [blockIdx.y * DIMQ + c] = ss;
}

__global__ void k_bnfin(const float* __restrict__ psum, const float* __restrict__ psq,
                        float* __restrict__ mean, float* __restrict__ rstd) {
  const int c = blockIdx.x * 256 + threadIdx.x;
  float s = 0.f, ss = 0.f;
#pragma unroll
  for (int i = 0; i < 16; ++i) { s += psum[i * DIMQ + c]; ss += psq[i * DIMQ + c]; }
  float m = s / (float)BT;
  float v = ss / (float)BT - m * m;
  mean[c] = m;
  rstd[c] = rsqrtf(v + 1e-5f);
}

// -------------------------- dots[bt,h,p,n] = sum_d qn[bt,p,h,d] * keys[h,n,p,d]
__global__ void k_dots(const float* __restrict__ q, const float* __restrict__ mean,
                       const float* __restrict__ rstd, const float* __restrict__ keys,
                       float* __restrict__ dots) {
  const int lane  = threadIdx.x & 31;
  const int wid   = threadIdx.x >> 5;
  const int combo = blockIdx.z;                   // 0..7 = h*2+p
  const int h = combo >> 1, p = combo & 1;
  const int m0 = blockIdx.y * 16;                 // token tile
  const int n0 = (blockIdx.x * 8 + wid) * 16;     // key tile (gridDim.x = 2)
  const int cbase = p * 512 + h * DH;
  const int r  = lane & 15;
  const int kh = (lane >> 4) * 2;
  const float* ap = q + (size_t)(m0 + r) * DIMQ + cbase + kh;
  const float* mp = mean + cbase + kh;
  const float* rp = rstd + cbase + kh;
  const float* bp = keys + (size_t)((h * NUM_KEYS + n0 + r) * 2 + p) * DH + kh;
  v8f c = {};
  for (int k = 0; k < DH; k += 4) {
    v2f a;
    a.x = (ap[0] - mp[0]) * rp[0];                // BN folded into A-load
    a.y = (ap[1] - mp[1]) * rp[1];
    v2f b = *(const v2f*)bp;
    c = wmma_f32_16x16x4(a, b, c);
    ap += 4; mp += 4; rp += 4; bp += 4;
  }
  const int crow = m0 + ((lane >> 4) << 3);
  const int col  = n0 + (lane & 15);
#pragma unroll
  for (int i = 0; i < 8; ++i)
    dots[(size_t)((crow + i) * HEADS + h) * (2 * NUM_KEYS) + p * NUM_KEYS + col] = c[i];
}

// ---------------------- per-(bt,h): top32(256)^2 -> top32(32x32) -> softmax/idx
__global__ void k_topk(const float* __restrict__ dots,
                       float* __restrict__ wout, int* __restrict__ iout) {
  __shared__ float s_sx[4][TOPK];
  __shared__ int   s_ix[4][TOPK];
  const int lane = threadIdx.x & 31;
  const int wid  = threadIdx.x >> 5;
  const int gw   = blockIdx.x * 4 + wid;          // 0..16383
  const int bt = gw >> 2;
  const int h  = gw & 3;

  // phase 1a: top-32 of 256 for p=0, results to LDS
  {
    const float* dp = dots + (size_t)((bt * HEADS + h) * 2 + 0) * NUM_KEYS;
    float v[8];
#pragma unroll
    for (int j = 0; j < 8; ++j) v[j] = dp[lane + 32 * j];
    for (int k = 0; k < TOPK; ++k) {
      float bv = v[0]; int bk = lane;
#pragma unroll
      for (int j = 1; j < 8; ++j)
        if (v[j] > bv) { bv = v[j]; bk = (j << 5) | lane; }
      wave_argmax(bv, bk);
      if (lane == (bk & 31)) v[bk >> 5] = -INFINITY;
      if (lane == k) { s_sx[wid][k] = bv; s_ix[wid][k] = bk; }
    }
  }
  // phase 1b: top-32 of 256 for p=1, lane j keeps the j-th best
  float sy = 0.f; int iy = 0;
  {
    const float* dp = dots + (size_t)((bt * HEADS + h) * 2 + 1) * NUM_KEYS;
    float v[8];
#pragma unroll
    for (int j = 0; j < 8; ++j) v[j] = dp[lane + 32 * j];
    for (int k = 0; k < TOPK; ++k) {
      float bv = v[0]; int bk = lane;
#pragma unroll
      for (int j = 1; j < 8; ++j)
        if (v[j] > bv) { bv = v[j]; bk = (j << 5) | lane; }
      wave_argmax(bv, bk);
      if (lane == (bk & 31)) v[bk >> 5] = -INFINITY;
      if (lane == k) { sy = bv; iy = bk; }
    }
  }
  __syncthreads();

  // phase 2: top-32 of the 32x32 sums; flat key (i<<5)|j matches jax position order
  unsigned alive = 0xFFFFFFFFu;                   // lane j owns column j, bits = rows i
  float myscore = 0.f; int myidx = 0;
  for (int k = 0; k < TOPK; ++k) {
    float bv = -INFINITY; int bk = 0x7FFFFFFF;
    for (int i = 0; i < 32; ++i) {
      if (alive & (1u << i)) {
        float s = s_sx[wid][i] + sy;
        int kk = (i << 5) | lane;
        if (s > bv || (s == bv && kk < bk)) { bv = s; bk = kk; }
      }
    }
    wave_argmax(bv, bk);
    const int istar = bk >> 5, jstar = bk & 31;
    if (lane == jstar) alive &= ~(1u << istar);
    const int iyj  = __shfl(iy, jstar, 32);
    const int cidx = s_ix[wid][istar] * NUM_KEYS + iyj;
    if (lane == k) { myscore = bv; myidx = cidx; }
  }

  // softmax over the 32 selected scores (descending, so lane 0 holds the max)
  const float mmax = __shfl(myscore, 0, 32);
  const float e = __expf(myscore - mmax);
  float ssum = e;
#pragma unroll
  for (int off = 16; off > 0; off >>= 1) ssum += __shfl_xor(ssum, off, 32);
  wout[(size_t)(bt * HEADS + h) * TOPK + lane] = e / ssum;
  iout[(size_t)(bt * HEADS + h) * TOPK + lane] = myidx;
}

// -------------------- out[bt,:] = sum_{h,k} w[h,k] * values_w[idx[h,k], :]
__global__ void k_out(const float* __restrict__ values_w, const float* __restrict__ wts,
                      const int* __restrict__ vidx, float* __restrict__ out) {
  __shared__ float lw[HEADS * TOPK];
  __shared__ int   li[HEADS * TOPK];
  const int bt = blockIdx.x;
  const int t  = threadIdx.x;                     // 0..127
  lw[t] = wts[(size_t)bt * (HEADS * TOPK) + t];
  li[t] = vidx[(size_t)bt * (HEADS * TOPK) + t];
  __syncthreads();
  const float4* vw = (const float4*)values_w;     // row = 128 x float4 (L2-resident)
  float4 acc = make_float4(0.f, 0.f, 0.f, 0.f);
#pragma unroll 4
  for (int k = 0; k < HEADS * TOPK; ++k) {
    const float w = lw[k];
    const float4 v = vw[(size_t)li[k] * (DIMX / 4) + t];
    acc.x += w * v.x; acc.y += w * v.y; acc.z += w * v.z; acc.w += w * v.w;
  }
  ((float4*)out)[(size_t)bt * (DIMX / 4) + t] = acc;
}

extern "C" void kernel_launch(void* const* d_in, const int* in_sizes, int n_in,
                              void* d_out, int out_size, void* d_ws, size_t ws_size,
                              hipStream_t stream) {
  const float* x    = (const float*)d_in[0];
  const float* wq   = (const float*)d_in[1];
  const float* keys = (const float*)d_in[2];
  const float* vwv  = (const float*)d_in[3];
  float* out = (float*)d_out;

  char* ws = (char*)d_ws;
  float* q    = (float*)ws; ws += (size_t)BT * DIMQ * sizeof(float);             // 16.8 MB
  float* psum = (float*)ws; ws += (size_t)16 * DIMQ * sizeof(float);
  float* psq  = (float*)ws; ws += (size_t)16 * DIMQ * sizeof(float);
  float* mean = (float*)ws; ws += (size_t)DIMQ * sizeof(float);
  float* rstd = (float*)ws; ws += (size_t)DIMQ * sizeof(float);
  float* dots = (float*)ws; ws += (size_t)BT * HEADS * 2 * NUM_KEYS * sizeof(float); // 33.6 MB
  float* wts  = (float*)ws; ws += (size_t)BT * HEADS * TOPK * sizeof(float);
  int*   vidx = (int*)  ws; ws += (size_t)BT * HEADS * TOPK * sizeof(int);

  k_qgemm <<<dim3(DIMQ / 128, BT / 16), 256, 0, stream>>>(x, wq, q);
  k_bnstat<<<dim3(DIMQ / 256, 16),      256, 0, stream>>>(q, psum, psq);
  k_bnfin <<<dim3(DIMQ / 256),          256, 0, stream>>>(psum, psq, mean, rstd);
  k_dots  <<<dim3(NUM_KEYS / 128, BT / 16, HEADS * 2), 256, 0, stream>>>(q, mean, rstd, keys, dots);
  k_topk  <<<dim3(BT), 128, 0, stream>>>(dots, wts, vidx);
  k_out   <<<dim3(BT), 128, 0, stream>>>(vwv, wts, vidx, out);
}